// CopyTemplate_87230785782203
// MI455X (gfx1250) — compile-verified
//
#include <hip/hip_runtime.h>

#define B_ 8
#define N_ 6
#define M_ 64
#define V_ 32000
#define T_ 8
#define S_ 7                      // N+1 template slots
#define MV ((size_t)M_ * V_)      // stride between s slices of input_decodings

typedef float v2f __attribute__((ext_vector_type(2)));
typedef float v8f __attribute__((ext_vector_type(8)));

// --------------------------------------------------------------------------
// One 16-column tile of out[t, v] for a fixed (b, m):
//   out = [16(t) x 4(k)] @ [4(k) x 16(v)]  twice (K = s'0..3, then s'4,5 + pad)
// WMMA #2 K-slot map: K0->s'=4, K1->s'=5, K2->pad choice (one-hot col 0 when
// m==0), K3->zero.  All loads unconditional; selects are cndmask (EXEC all-1s).
// --------------------------------------------------------------------------
__device__ __forceinline__ v8f tile_mma(const float* __restrict__ base, int col,
                                        bool hi, int m, v2f A1, v2f A2, int sB1) {
  float b10 = base[(size_t)sB1 * MV + col];
  float b11 = base[(size_t)(sB1 + 1) * MV + col];
  float x4  = base[(size_t)4 * MV + col];
  float x5  = base[(size_t)5 * MV + col];
  float padb = (m == 0 && col == 0) ? 1.0f : 0.0f;  // B[K=2, col] one-hot
  float b20 = hi ? padb : x4;
  float b21 = hi ? 0.0f : x5;
  v2f B1 = {b10, b11};
  v2f B2 = {b20, b21};
  v8f c = {0.f, 0.f, 0.f, 0.f, 0.f, 0.f, 0.f, 0.f};
  c = __builtin_amdgcn_wmma_f32_16x16x4_f32(false, A1, false, B1, (short)0, c,
                                            false, false);
  c = __builtin_amdgcn_wmma_f32_16x16x4_f32(false, A2, false, B2, (short)0, c,
                                            false, false);
  return c;
}

// --------------------------------------------------------------------------
// Kernel A: flag[b,t,m] = (out[b,t,m,0] >= max_v out[b,t,m,v])  (== argmax==0)
// Block = 256 threads (8 waves), one block per (m, b); 2000 tiles of 16 cols.
// --------------------------------------------------------------------------
__global__ __launch_bounds__(256)
void flags_kernel(const float* __restrict__ input, const float* __restrict__ tmpl,
                  const int* __restrict__ spans, int* __restrict__ flags) {
  const int m    = blockIdx.x;
  const int b    = blockIdx.y;
  const int tid  = threadIdx.x;
  const int lane = tid & 31;
  const int wave = tid >> 5;
  const int trow = lane & 15;      // A row (t) / B column-in-tile
  const bool hi  = lane >= 16;

  __shared__ float sA[T_][S_];     // span-masked template coefficients
  __shared__ float sOut0[T_];      // out[t, v=0] (pad included)
  __shared__ float sWMax[8][T_];   // per-wave running max per t

  if (tid < T_ * S_) {
    int t = tid / S_, s = tid % S_;
    float v = tmpl[(b * T_ + t) * S_ + s];
    sA[t][s] = (s <= spans[b]) ? v : 0.0f;
  }
  __syncthreads();

  // A layout (f32 16x4): VGPR0 -> K = hi?2:0, VGPR1 -> K = hi?3:1
  // WMMA #1: K -> choices s = K+1
  float a10 = (trow < T_) ? sA[trow][(hi ? 2 : 0) + 1] : 0.0f;
  float a11 = (trow < T_) ? sA[trow][(hi ? 2 : 0) + 2] : 0.0f;
  // WMMA #2: lanes<16 hold K0,K1 -> s=5,6 ; lanes>=16 hold K2 -> pad coeff, K3 -> 0
  float a20 = (trow < T_) ? (hi ? sA[trow][0] : sA[trow][5]) : 0.0f;
  float a21 = (!hi && trow < T_) ? sA[trow][6] : 0.0f;
  v2f A1 = {a10, a11};
  v2f A2 = {a20, a21};

  const float* base = input + (size_t)b * N_ * MV + (size_t)m * V_;
  const int sB1  = hi ? 2 : 0;       // B rows for WMMA #1
  const int off0 = wave * 16 + trow; // column for i==0

  float rmax[T_];
#pragma unroll
  for (int g = 0; g < T_; ++g) rmax[g] = -__builtin_inff();

  // Peeled first iteration (wave 0 owns column v=0): capture out0 from the
  // exact WMMA result so the flag compare is bit-consistent with rmax.
  {
    v8f c = tile_mma(base, off0, hi, m, A1, A2, sB1);
    if (wave == 0 && lane == 0) {
#pragma unroll
      for (int g = 0; g < T_; ++g) sOut0[g] = c[g];
    }
#pragma unroll
    for (int g = 0; g < T_; ++g) rmax[g] = fmaxf(rmax[g], c[g]);
  }
  // Steady-state: branch-free tile loop (250 tiles/wave, waves interleaved).
  for (int i = 1; i < (V_ / 16) / 8; ++i) {
    v8f c = tile_mma(base, i * 128 + off0, hi, m, A1, A2, sB1);
#pragma unroll
    for (int g = 0; g < T_; ++g) rmax[g] = fmaxf(rmax[g], c[g]);
  }

  // Reduce max across lanes 0..15 (xor<16 keeps lane halves separate);
  // lanes 16-31 carry rows t=8..15 (all-zero A rows) and are ignored.
#pragma unroll
  for (int g = 0; g < T_; ++g) {
    float v = rmax[g];
    v = fmaxf(v, __shfl_xor(v, 1, 32));
    v = fmaxf(v, __shfl_xor(v, 2, 32));
    v = fmaxf(v, __shfl_xor(v, 4, 32));
    v = fmaxf(v, __shfl_xor(v, 8, 32));
    rmax[g] = v;
  }
  if (lane == 0) {
#pragma unroll
    for (int g = 0; g < T_; ++g) sWMax[wave][g] = rmax[g];
  }
  __syncthreads();
  if (tid < T_) {
    float mx = sWMax[0][tid];
#pragma unroll
    for (int w = 1; w < 8; ++w) mx = fmaxf(mx, sWMax[w][tid]);
    flags[(b * T_ + tid) * M_ + m] = (sOut0[tid] >= mx) ? 1 : 0;
  }
}

// --------------------------------------------------------------------------
// Kernel B: serial replay of the scan, per batch. Emits, for every output row
// m, the source (t, j) it copies from, or t = -1 if it stays zero.
// --------------------------------------------------------------------------
__global__ void plan_kernel(const int* __restrict__ flags, int* __restrict__ srcT,
                            int* __restrict__ srcJ) {
  int b = threadIdx.x;
  if (b >= B_) return;
  int idx = 0;
  for (int t = 0; t < T_; ++t) {
    int len = M_;
    for (int mm = 0; mm < M_; ++mm) {
      if (flags[(b * T_ + t) * M_ + mm]) { len = mm; break; }
    }
    int rem = M_ - idx;
    if (len > rem) len = rem;
    for (int j = 0; j < len; ++j) {
      srcT[b * M_ + idx + j] = t;
      srcJ[b * M_ + idx + j] = j;
    }
    idx += len;
  }
  for (int mm = idx; mm < M_; ++mm) srcT[b * M_ + mm] = -1;
}

// --------------------------------------------------------------------------
// Kernel C: materialize result rows. One block per (m, b); recompute the
// selected out row (coalesced) or write zeros.
// --------------------------------------------------------------------------
__global__ __launch_bounds__(256)
void write_kernel(const float* __restrict__ input, const float* __restrict__ tmpl,
                  const int* __restrict__ spans, const int* __restrict__ srcT,
                  const int* __restrict__ srcJ, float* __restrict__ out) {
  const int m   = blockIdx.x;
  const int b   = blockIdx.y;
  const int tid = threadIdx.x;
  float* orow = out + ((size_t)b * M_ + m) * V_;

  const int t = srcT[b * M_ + m];
  if (t < 0) {
    for (int v = tid; v < V_; v += 256) orow[v] = 0.0f;
    return;
  }
  const int j    = srcJ[b * M_ + m];
  const int span = spans[b];

  float w[N_];
#pragma unroll
  for (int s = 0; s < N_; ++s)
    w[s] = (s < span) ? tmpl[(b * T_ + t) * S_ + s + 1] : 0.0f;
  const float w0 = tmpl[(b * T_ + t) * S_ + 0];

  const float* base = input + (size_t)b * N_ * MV + (size_t)j * V_;
  for (int v = tid; v < V_; v += 256) {
    float acc = 0.0f;
#pragma unroll
    for (int s = 0; s < N_; ++s) acc = fmaf(w[s], base[(size_t)s * MV + v], acc);
    if (j == 0 && v == 0) acc += w0;   // pad one-hot lands at (m_src=0, v=0)
    orow[v] = acc;
  }
}

// --------------------------------------------------------------------------
extern "C" void kernel_launch(void* const* d_in, const int* in_sizes, int n_in,
                              void* d_out, int out_size, void* d_ws, size_t ws_size,
                              hipStream_t stream) {
  (void)in_sizes; (void)n_in; (void)out_size; (void)ws_size;
  const float* input = (const float*)d_in[0];   // [B,N,M,V] f32
  const float* tmpl  = (const float*)d_in[1];   // [B,T,N+1] f32
  const int*   spans = (const int*)d_in[2];     // [B] i32
  float* out = (float*)d_out;                   // [B,M,V] f32

  int* flags = (int*)d_ws;                      // B*T*M ints
  int* srcT  = flags + B_ * T_ * M_;            // B*M ints
  int* srcJ  = srcT + B_ * M_;                  // B*M ints

  flags_kernel<<<dim3(M_, B_), 256, 0, stream>>>(input, tmpl, spans, flags);
  plan_kernel<<<1, 32, 0, stream>>>(flags, srcT, srcJ);
  write_kernel<<<dim3(M_, B_), 256, 0, stream>>>(input, tmpl, spans, srcT, srcJ, out);
}